// GCN_84086869721211
// MI455X (gfx1250) — compile-verified
//
#include <hip/hip_runtime.h>
#include <hip/hip_bf16.h>

// CDNA5 / gfx1250: wave32, WMMA (not MFMA).
typedef float v2f __attribute__((ext_vector_type(2)));
typedef float v8f __attribute__((ext_vector_type(8)));

#define NFEAT 128
#define ODIM  64

// ---------------------------------------------------------------------------
// Degree / normalization kernels
// ---------------------------------------------------------------------------
__global__ void GCN_deg_init_kernel(float* __restrict__ deg, int n) {
    int i = blockIdx.x * blockDim.x + threadIdx.x;
    if (i < n) deg[i] = 1.0f;  // self-loop contributes 1 to every target degree
}

__global__ void GCN_deg_count_kernel(const int* __restrict__ dst,
                                     float* __restrict__ deg, int E) {
    int e = blockIdx.x * blockDim.x + threadIdx.x;
    if (e < E) unsafeAtomicAdd(&deg[dst[e]], 1.0f);  // global_atomic_add_f32
}

__global__ void GCN_dinv_kernel(float* __restrict__ deg, int n) {
    int i = blockIdx.x * blockDim.x + threadIdx.x;
    if (i < n) {
        float d = deg[i];
        deg[i] = (d > 0.0f) ? rsqrtf(d) : 0.0f;  // deg >= 1 here, but keep guard
    }
}

// ---------------------------------------------------------------------------
// GEMM: C[M x 64] = (relu?)(A[M x K]) @ B[K x 64]   via V_WMMA_F32_16X16X4_F32
// One wave (32 lanes) computes a 16x64 stripe of C: a single 16x4 f32 A
// fragment is reused across the four 16-wide N tiles (4 accumulators).
//
// A fragment layout (ISA 7.12.2, 32-bit A 16x4): lanes 0-15 hold M=lane,
// K={k0,k0+1}; lanes 16-31 hold M=lane-16, K={k0+2,k0+3}.
// B (4x16) striped by K across the two half-waves, N across lanes.
// C/D 16x16 f32: VGPR r -> M = r (lanes 0-15) / r+8 (lanes 16-31), N = lane&15.
// ---------------------------------------------------------------------------
__global__ __launch_bounds__(32)
void GCN_gemm_wmma_kernel(const float* __restrict__ A,
                          const float* __restrict__ B,
                          float* __restrict__ C,
                          int M, int K, int relu_in) {
    const int lane = threadIdx.x;
    const int half = lane >> 4;     // 0: lanes 0-15, 1: lanes 16-31
    const int l    = lane & 15;
    const int row0 = blockIdx.x * 16;

    int rowA = row0 + l;
    if (rowA >= M) rowA = M - 1;    // clamp (M is a multiple of 16 here anyway)
    const long long abase = (long long)rowA * K;

    v8f acc[ODIM / 16];
#pragma unroll
    for (int t = 0; t < ODIM / 16; ++t)
        acc[t] = (v8f){0.f, 0.f, 0.f, 0.f, 0.f, 0.f, 0.f, 0.f};

    for (int k0 = 0; k0 < K; k0 += 4) {
        const int ka = k0 + half * 2;
        v2f a;
        a.x = A[abase + ka];
        a.y = A[abase + ka + 1];
        if (relu_in) {              // fused layer-1 ReLU on the fly
            a.x = fmaxf(a.x, 0.0f);
            a.y = fmaxf(a.y, 0.0f);
        }
#pragma unroll
        for (int t = 0; t < ODIM / 16; ++t) {
            const int col = t * 16 + l;
            v2f b;
            b.x = B[ka * ODIM + col];
            b.y = B[(ka + 1) * ODIM + col];
            // 8 args: (neg_a, A, neg_b, B, c_mod, C, reuse_a, reuse_b)
            acc[t] = __builtin_amdgcn_wmma_f32_16x16x4_f32(
                false, a, false, b, (short)0, acc[t], false, false);
        }
    }

#pragma unroll
    for (int t = 0; t < ODIM / 16; ++t) {
        const int col = t * 16 + l;
#pragma unroll
        for (int r = 0; r < 8; ++r) {
            const int row = row0 + half * 8 + r;
            if (row < M) C[(long long)row * ODIM + col] = acc[t][r];
        }
    }
}

// ---------------------------------------------------------------------------
// out[i,d] = h[i,d] * dinv[i]^2 + bias[d]   (self-loop term + bias, also
// fully initializes the destination buffer each call)
// ---------------------------------------------------------------------------
__global__ void GCN_selfloop_bias_kernel(const float* __restrict__ h,
                                         const float* __restrict__ dinv,
                                         const float* __restrict__ bias,
                                         float* __restrict__ out,
                                         long long total) {
    long long t = (long long)blockIdx.x * blockDim.x + threadIdx.x;
    if (t < total) {
        long long i = t >> 6;            // node
        int d = (int)(t & 63);           // feature
        float w = dinv[i];
        out[t] = h[t] * w * w + bias[d];
    }
}

// ---------------------------------------------------------------------------
// Edge scatter: out[dst] += h[src] * dinv[src]*dinv[dst]
// One thread per (edge, feature): consecutive lanes touch consecutive
// features of the same edge -> coalesced 128B gathers per wave, native
// f32 atomics to the destination rows (working set fits in the 192MB L2).
// ---------------------------------------------------------------------------
__global__ void GCN_scatter_kernel(const float* __restrict__ h,
                                   const int* __restrict__ src,
                                   const int* __restrict__ dst,
                                   const float* __restrict__ dinv,
                                   float* __restrict__ out,
                                   long long E) {
    const long long total  = E << 6;     // E * 64 features
    const long long stride = (long long)gridDim.x * blockDim.x;
    for (long long t = (long long)blockIdx.x * blockDim.x + threadIdx.x;
         t < total; t += stride) {
        const long long e = t >> 6;
        const int d = (int)(t & 63);
        const int s = src[e];
        const int v = dst[e];
        const float norm = dinv[s] * dinv[v];
        unsafeAtomicAdd(&out[(long long)v * ODIM + d],
                        h[(long long)s * ODIM + d] * norm);
    }
}

// ---------------------------------------------------------------------------
extern "C" void kernel_launch(void* const* d_in, const int* in_sizes, int n_in,
                              void* d_out, int out_size, void* d_ws, size_t ws_size,
                              hipStream_t stream) {
    const float* x  = (const float*)d_in[0];
    const int*   ei = (const int*)d_in[1];   // [2, E] flat: row 0 = src, row 1 = dst
    const float* W1 = (const float*)d_in[2];
    const float* b1 = (const float*)d_in[3];
    const float* W2 = (const float*)d_in[4];
    const float* b2 = (const float*)d_in[5];

    const int       n = in_sizes[0] / NFEAT;   // 100000
    const long long E = in_sizes[1] / 2;       // 3200000
    const int*  src = ei;
    const int*  dst = ei + E;

    // Workspace layout (floats): dinv[n] | h0[n*64] (reused as h2) | h1[n*64]
    float* ws     = (float*)d_ws;
    float* d_dinv = ws;
    float* h0     = ws + n;
    float* h1     = h0 + (long long)n * ODIM;
    float* out    = (float*)d_out;

    const long long nd   = (long long)n * ODIM;
    const int  TB        = 256;
    const int  nBlocksN  = (n + TB - 1) / TB;
    const int  nBlocksE  = (int)((E + TB - 1) / TB);
    const int  nBlocksND = (int)((nd + TB - 1) / TB);
    const int  nBlocksEd = (int)(((E << 6) + TB - 1) / TB);
    const int  mTiles    = (n + 15) / 16;

    // Degree -> dinv
    GCN_deg_init_kernel<<<nBlocksN, TB, 0, stream>>>(d_dinv, n);
    GCN_deg_count_kernel<<<nBlocksE, TB, 0, stream>>>(dst, d_dinv, (int)E);
    GCN_dinv_kernel<<<nBlocksN, TB, 0, stream>>>(d_dinv, n);

    // Layer 1: h0 = x @ W1 ; h1 = segsum + selfloop + b1 (ReLU deferred to GEMM2 load)
    GCN_gemm_wmma_kernel<<<mTiles, 32, 0, stream>>>(x, W1, h0, n, NFEAT, 0);
    GCN_selfloop_bias_kernel<<<nBlocksND, TB, 0, stream>>>(h0, d_dinv, b1, h1, nd);
    GCN_scatter_kernel<<<nBlocksEd, TB, 0, stream>>>(h0, src, dst, d_dinv, h1, E);

    // Layer 2: h2 = relu(h1) @ W2 (h2 reuses h0's buffer) ; out = segsum + selfloop + b2
    GCN_gemm_wmma_kernel<<<mTiles, 32, 0, stream>>>(h1, W2, h0, n, ODIM, 1);
    GCN_selfloop_bias_kernel<<<nBlocksND, TB, 0, stream>>>(h0, d_dinv, b2, out, nd);
    GCN_scatter_kernel<<<nBlocksEd, TB, 0, stream>>>(h0, src, dst, d_dinv, out, E);
}